// FPS_65773129171297
// MI455X (gfx1250) — compile-verified
//
#include <hip/hip_runtime.h>
#include <stdint.h>

// Problem constants (from reference)
#define FPS_B 32
#define FPS_N 262144
#define FPS_S 2048          // NUM_POINTS
#define BPB   8             // blocks per batch
#define NTHREADS 256        // 8 wave32 waves per block
#define NWAVES (NTHREADS / 32)
#define SLICE (FPS_N / BPB)             // 32768 points per block
#define CHUNK_PTS 2048                  // points per LDS chunk
#define NCHUNK (SLICE / CHUNK_PTS)      // 16 chunks per iteration
#define CHUNK_BYTES (CHUNK_PTS * 12)    // 24576 B

// Low 32 bits of a flat pointer into __shared__ == LDS offset (shared aperture
// has zero low bits; HW adds the wave's LDS_BASE).
__device__ __forceinline__ uint32_t lds_lo32(const void* p) {
  return (uint32_t)(uintptr_t)p;
}

// CDNA5 async global->LDS copy (16B per lane). VDST = LDS byte address VGPR,
// VADDR = 64-bit global address, saddr = off. Tracked by per-wave ASYNCcnt.
__device__ __forceinline__ void async_b128(uint32_t lds, unsigned long long gaddr) {
  asm volatile("global_load_async_to_lds_b128 %0, %1, off"
               :: "v"(lds), "v"(gaddr)
               : "memory");
}

__device__ __forceinline__ void wait_async_le6() {
#if __has_builtin(__builtin_amdgcn_s_wait_asynccnt)
  __builtin_amdgcn_s_wait_asynccnt(6);
#else
  asm volatile("s_wait_asynccnt 0x6" ::: "memory");
#endif
}
__device__ __forceinline__ void wait_async_0() {
#if __has_builtin(__builtin_amdgcn_s_wait_asynccnt)
  __builtin_amdgcn_s_wait_asynccnt(0);
#else
  asm volatile("s_wait_asynccnt 0x0" ::: "memory");
#endif
}

// 64-bit wave32 xor-butterfly max (two 32-bit shuffles per step)
__device__ __forceinline__ unsigned long long shfl_xor_u64(unsigned long long v,
                                                           int m) {
  uint32_t lo = (uint32_t)v, hi = (uint32_t)(v >> 32);
  lo = __shfl_xor((unsigned int)lo, m, 32);
  hi = __shfl_xor((unsigned int)hi, m, 32);
  return ((unsigned long long)hi << 32) | lo;
}

// ---------------------------------------------------------------------------
// Init: min_dist = +inf, zero per-iteration argmax slots / arrive counters,
// and emit the preset sample 0 (point 0, index 0) for each batch.
// ---------------------------------------------------------------------------
__global__ void fps_init(const float* __restrict__ points, float* __restrict__ out,
                         float* __restrict__ md, unsigned long long* __restrict__ slot,
                         unsigned int* __restrict__ cnt) {
  size_t tid = (size_t)blockIdx.x * blockDim.x + threadIdx.x;
  if (tid < (size_t)FPS_B * FPS_N) md[tid] = __builtin_huge_valf();
  if (tid < (size_t)FPS_S * FPS_B) { slot[tid] = 0ull; cnt[tid] = 0u; }
  if (tid < FPS_B) {
    const float* p = points + tid * (size_t)FPS_N * 3;
    float* os = out + tid * (size_t)FPS_S * 3;
    os[0] = p[0]; os[1] = p[1]; os[2] = p[2];
    out[(size_t)FPS_B * FPS_S * 3 + tid * (size_t)FPS_S] = 0.0f;  // idx 0
  }
}

// ---------------------------------------------------------------------------
// Persistent cooperative FPS kernel: 32 batches x 8 blocks x 256 threads.
// Each lane DMAs exactly the bytes it later reads -> no block barriers in the
// chunk pipeline (per-wave ASYNCcnt + in-order async completion suffice).
// ---------------------------------------------------------------------------
__global__ __launch_bounds__(NTHREADS) void fps_main(
    const float* __restrict__ points, float* __restrict__ out,
    float* __restrict__ md_all, unsigned long long* __restrict__ slot,
    unsigned int* __restrict__ cnt) {
  __shared__ __align__(16) float buf[2][CHUNK_PTS * 3];  // 48 KB double buffer
  __shared__ unsigned long long red[NWAVES];
  __shared__ unsigned long long sh_win;

  const int t  = threadIdx.x;
  const int b  = blockIdx.x / BPB;   // batch
  const int bb = blockIdx.x % BPB;   // block-in-batch
  const size_t batch_base   = (size_t)b * FPS_N;
  const uint32_t slice_base = (uint32_t)bb * SLICE;
  const float* slicep = points + (batch_base + (size_t)slice_base) * 3;
  float* md = md_all + batch_base + slice_base;  // block-private slice of min_dist

  // This lane's two 4-point groups within a chunk: byte offsets of 48 B each.
  const uint32_t own_off0 = (uint32_t)t * 48u;            // points [4t, 4t+4)
  const uint32_t own_off1 = own_off0 + 1024u * 12u;       // points [1024+4t, ...)

  uint32_t prev_idx = 0;  // sample 0 is point 0

  for (int i = 1; i < FPS_S; ++i) {
    const float* q = points + (batch_base + prev_idx) * 3;
    float qx = q[0], qy = q[1], qz = q[2];

    float bestv = -1.0f;
    uint32_t besti = 0;

    // --- async DMA issue for one chunk: 6 x b128 = this lane's own 96 B ----
    auto issue_chunk = [&](int c, int par) {
      const unsigned long long gbase = (unsigned long long)(uintptr_t)slicep +
                                       (unsigned long long)c * CHUNK_BYTES;
      const uint32_t lbase = lds_lo32(&buf[par][0]);
#pragma unroll
      for (int j = 0; j < 3; ++j)
        async_b128(lbase + own_off0 + j * 16u, gbase + own_off0 + j * 16u);
#pragma unroll
      for (int j = 0; j < 3; ++j)
        async_b128(lbase + own_off1 + j * 16u, gbase + own_off1 + j * 16u);
    };

    issue_chunk(0, 0);

    for (int c = 0; c < NCHUNK; ++c) {
      const int par = c & 1;
      if (c + 1 < NCHUNK) {
        issue_chunk(c + 1, par ^ 1);  // overlap next chunk's DMA with compute
        wait_async_le6();             // in-order => this lane's chunk c is in LDS
      } else {
        wait_async_0();
      }

      const float* lb = &buf[par][0];
      const int cbase = c * CHUNK_PTS;
#pragma unroll
      for (int m = 0; m < 2; ++m) {
        const int p0 = (t << 2) + (m << 10);  // 4t (+1024)
        const float4* lv = (const float4*)(lb + (size_t)p0 * 3);
        float4 v0 = lv[0], v1 = lv[1], v2 = lv[2];  // 4 xyz points, AoS
        const int gi = cbase + p0;
        float4 mo = *(const float4*)(md + gi);
        float dx, dy, dz, d;
        float4 mn;
        dx = v0.x - qx; dy = v0.y - qy; dz = v0.z - qz;
        d = fmaf(dx, dx, fmaf(dy, dy, dz * dz)); mn.x = fminf(mo.x, d);
        dx = v0.w - qx; dy = v1.x - qy; dz = v1.y - qz;
        d = fmaf(dx, dx, fmaf(dy, dy, dz * dz)); mn.y = fminf(mo.y, d);
        dx = v1.z - qx; dy = v1.w - qy; dz = v2.x - qz;
        d = fmaf(dx, dx, fmaf(dy, dy, dz * dz)); mn.z = fminf(mo.z, d);
        dx = v2.y - qx; dy = v2.z - qy; dz = v2.w - qz;
        d = fmaf(dx, dx, fmaf(dy, dy, dz * dz)); mn.w = fminf(mo.w, d);
        *(float4*)(md + gi) = mn;
        const uint32_t gidx = slice_base + (uint32_t)gi;
        if (mn.x > bestv) { bestv = mn.x; besti = gidx;      }
        if (mn.y > bestv) { bestv = mn.y; besti = gidx + 1u; }
        if (mn.z > bestv) { bestv = mn.z; besti = gidx + 2u; }
        if (mn.w > bestv) { bestv = mn.w; besti = gidx + 3u; }
      }
      // no __syncthreads(): each wave reads only LDS it DMA'd itself
    }

    // --- argmax reduction: packed key, wave32 butterfly then 8-wave LDS ----
    unsigned long long key =
        ((unsigned long long)__float_as_uint(bestv) << 32) |
        (unsigned long long)(uint32_t)(~besti);
#pragma unroll
    for (int mdist = 16; mdist > 0; mdist >>= 1) {
      unsigned long long o = shfl_xor_u64(key, mdist);
      if (o > key) key = o;
    }
    if ((t & 31) == 0) red[t >> 5] = key;
    __syncthreads();

    const size_t sidx = (size_t)i * FPS_B + b;  // per-(iter,batch) slot
    if (t == 0) {
      unsigned long long k = red[0];
#pragma unroll
      for (int w = 1; w < NWAVES; ++w)
        if (red[w] > k) k = red[w];
      atomicMax(&slot[sidx], k);
      __threadfence();                          // max visible before arrive
      atomicAdd(&cnt[sidx], 1u);
      while (__hip_atomic_load(&cnt[sidx], __ATOMIC_ACQUIRE,
                               __HIP_MEMORY_SCOPE_AGENT) < (unsigned)BPB) {
        __builtin_amdgcn_s_sleep(2);
      }
      sh_win = atomicAdd(&slot[sidx], 0ull);    // coherent read of winner
    }
    __syncthreads();
    unsigned long long win = sh_win;
    uint32_t idx = ~(uint32_t)(win & 0xffffffffull);
    prev_idx = idx;

    if (bb == 0 && t == 0) {
      const float* p = points + (batch_base + idx) * 3;
      float* os = out + ((size_t)b * FPS_S + i) * 3;
      os[0] = p[0]; os[1] = p[1]; os[2] = p[2];
      out[(size_t)FPS_B * FPS_S * 3 + (size_t)b * FPS_S + i] = (float)idx;
    }
    __syncthreads();  // sh_win / red safe for reuse next iteration
  }
}

// ---------------------------------------------------------------------------
// Workspace layout (bytes):
//   [0, 32MB)          min_dist (B*N floats)
//   [32MB, +512KB)     slot     (S*B u64 packed argmax)
//   [+512KB, +768KB)   cnt      (S*B u32 arrive counters)
// Total ~34.3 MB (assumes ws_size is at least that).
// ---------------------------------------------------------------------------
extern "C" void kernel_launch(void* const* d_in, const int* in_sizes, int n_in,
                              void* d_out, int out_size, void* d_ws, size_t ws_size,
                              hipStream_t stream) {
  (void)in_sizes; (void)n_in; (void)out_size; (void)ws_size;
  const float* points = (const float*)d_in[0];
  float* out = (float*)d_out;

  char* ws = (char*)d_ws;
  float* md = (float*)ws;
  unsigned long long* slot =
      (unsigned long long*)(ws + (size_t)FPS_B * FPS_N * sizeof(float));
  unsigned int* cnt =
      (unsigned int*)(ws + (size_t)FPS_B * FPS_N * sizeof(float) +
                      (size_t)FPS_S * FPS_B * sizeof(unsigned long long));

  const int initBlocks = (FPS_B * FPS_N + NTHREADS - 1) / NTHREADS;
  fps_init<<<initBlocks, NTHREADS, 0, stream>>>(points, out, md, slot, cnt);
  fps_main<<<FPS_B * BPB, NTHREADS, 0, stream>>>(points, out, md, slot, cnt);
}